// MCPBRNN_PETconstraint_MassRelax_Indepedent_53695681134744
// MI455X (gfx1250) — compile-verified
//
#include <hip/hip_runtime.h>

// ---- reference constants ----
#define ML_C       2.9086f
#define SL_C       1.898f
#define SCALE_MR_C 500.0f
#define SPIN_C     1000
#define TRAIN_C    400000

#if __has_builtin(__builtin_amdgcn_tanhf)
#define HAS_TANH 1
#else
#define HAS_TANH 0
#endif

// sigmoid: gfx1250 v_tanh_f32 path (1 transcendental) with exp2/rcp fallback
__device__ __forceinline__ float fast_sigmoid(float z) {
#if HAS_TANH
    return fmaf(__builtin_amdgcn_tanhf(0.5f * z), 0.5f, 0.5f);
#else
    float e = __builtin_amdgcn_exp2f(z * -1.44269504088896340736f);
    return __builtin_amdgcn_rcpf(1.0f + e);
#endif
}

// uniform broadcast from lane k via v_readlane_b32 (SGPR result, no LDS)
__device__ __forceinline__ float bcast(float v, int k) {
    return __int_as_float(__builtin_amdgcn_readlane(__float_as_int(v), k));
}

// ---------------------------------------------------------------------------
// Kernel A: std(y_obs[SPIN:TRAIN], ddof=1) + parameter transforms ->
// ws[0..10]: 0=mo 1=so 2=wb1 3=wb2 4=b0_yom 5=b0_ylm 6=thr 7=oo1 8=ol1
//            9=svm 10=obsstd
// ---------------------------------------------------------------------------
__global__ __launch_bounds__(256) void k_stats(
    const float* __restrict__ y, int ny,
    const float* __restrict__ p_mean, const float* __restrict__ p_std,
    const float* __restrict__ w_r_yom, const float* __restrict__ w_r_ylm,
    const float* __restrict__ w_r_yfm, const float* __restrict__ w_r_yvm,
    const float* __restrict__ b0_yom, const float* __restrict__ w_b1_yom,
    const float* __restrict__ b0_ylm, const float* __restrict__ w_b2_ylm,
    const float* __restrict__ b0_yrm, float* __restrict__ ws)
{
    __shared__ double ss[256];
    __shared__ double sq[256];
    const int tid = threadIdx.x;
    const int lo = SPIN_C;
    const int hi = (ny < TRAIN_C) ? ny : TRAIN_C;

    double s = 0.0, q = 0.0;
    for (int i = lo + tid; i < hi; i += 256) {
        double v = (double)y[i];
        s += v;
        q += v * v;
    }
    ss[tid] = s; sq[tid] = q;
    __syncthreads();
    for (int o = 128; o > 0; o >>= 1) {
        if (tid < o) { ss[tid] += ss[tid + o]; sq[tid] += sq[tid + o]; }
        __syncthreads();
    }
    if (tid == 0) {
        double cnt = (double)(hi - lo);
        double var = (sq[0] - ss[0] * ss[0] / cnt) / (cnt - 1.0);
        if (!(var > 0.0)) var = 0.0;

        float eo  = __expf(w_r_yom[0]);
        float el  = __expf(w_r_ylm[0]);
        float ef  = __expf(w_r_yfm[0]);
        float den = eo + el + ef;

        ws[0]  = p_mean[0];
        ws[1]  = p_std[0];
        ws[2]  = w_b1_yom[0];
        ws[3]  = w_b2_ylm[0];
        ws[4]  = b0_yom[0];
        ws[5]  = b0_ylm[0];
        ws[6]  = __expf(b0_yrm[0]);        // thr
        ws[7]  = eo / den;                 // oo1
        ws[8]  = el / den;                 // ol1
        ws[9]  = fast_sigmoid(w_r_yvm[0]); // svm
        ws[10] = (float)sqrt(var);         // obsstd
    }
}

// ---------------------------------------------------------------------------
// Kernel B: the sequential recurrence on ONE wave32.
// Per 32-step tile: coalesced float2 loads of x, lane-parallel precompute of
// the u2-only gate `ol`, per-step operands pulled from lane k via
// v_readlane (SGPR path), pre-update state captured per lane, one coalesced
// 128B store per tile. Chain math minimized:
//   olc = pos ? min(ol, u2*rcp(c)) : ol          (== ol - relu(ol - u2/c))
//   f   = fma(-C1, tanh(fma(c,A1,B1)), E1) - olc (== 1 - oo - olc)
//   ov  = min(ov1, f)                            (== ov1 - relu(ov1 - f))
//   c'  = fma(-ov, |c - thrS|, fma(f, c, u1))
// ---------------------------------------------------------------------------
__global__ __launch_bounds__(32) void k_scan(
    const float* __restrict__ x, const float* __restrict__ ws,
    float* __restrict__ cstate, int n, const int* __restrict__ time_lag_p)
{
    const int lane = threadIdx.x;
    const int tl   = *time_lag_p;

    const float mo = ws[0], so = ws[1], wb1 = ws[2], wb2 = ws[3];
    const float b0_yom = ws[4], b0_ylm = ws[5], thr = ws[6];
    const float oo1 = ws[7], ol1 = ws[8], svm = ws[9];

    const float k1   = wb1 / so;
    const float k2   = wb2 * (1.0f / SL_C);
    const float thrS = thr * SCALE_MR_C;
    const float invS = 1.0f / SCALE_MR_C;
#if HAS_TANH
    const float A1 = 0.5f * k1;                    // half_z = fma(c, A1, B1)
    const float B1 = 0.5f * fmaf(-mo, k1, b0_yom);
    const float C1 = 0.5f * oo1;
    const float E1 = 1.0f - C1;                    // g = fma(-C1, t, E1)
#endif

    const int nTiles = (n + 31) >> 5;
    int tlc = tl < 0 ? 0 : (tl > n ? n : tl);
    const int t0 = tlc >> 5;          // tiles fully before time_lag: don't-care
    if (t0 >= nTiles) return;

    auto loadTile = [&](int t, float& a, float& b, float& o) {
        int idx = (t << 5) + lane;
        float va = 0.0f, vb = 0.0f;
        if (idx < n) {
            float2 u = reinterpret_cast<const float2*>(x)[idx];
            va = u.x; vb = u.y;
        }
        a = va; b = vb;
        o = ol1 * fast_sigmoid(fmaf(vb - ML_C, k2, b0_ylm)); // ol: u2-only
    };

    float c = 0.0f;
    float aC, bC, oC, aN, bN, oN;
    loadTile(t0, aC, bC, oC);

    for (int t = t0; t < nTiles; ++t) {
        if (t + 1 < nTiles) loadTile(t + 1, aN, bN, oN);  // pipelined next tile
        {
            int pidx = ((t + 16) << 5) + lane;            // gfx1250 global_prefetch
            if (pidx >= n) pidx = n - 1;
            __builtin_prefetch(x + 2 * pidx, 0, 3);
        }

        const int base = t << 5;
        const int cnt  = (n - base) < 32 ? (n - base) : 32;
        float cs = c;

        // one recurrence step; `check` selects the partially-active variant
        auto step = [&](int k, bool check) {
            float u1  = bcast(aC, k);
            float u2  = bcast(bC, k);
            float olk = bcast(oC, k);
            cs = (lane == k) ? c : cs;                    // capture pre-update c0
            if (!check || (base + k >= tl)) {
                // off-chain (depend only on c): am, sg, ov1, tanh side-path
                float am  = fabsf(c - thrS);
                float d   = c * invS - thr;
                float sg  = (d > 0.0f) ? 1.0f : ((d < 0.0f) ? -1.0f : 0.0f);
                float ov1 = svm * sg;
#if HAS_TANH
                float tt = __builtin_amdgcn_tanhf(fmaf(c, A1, B1));
                float g  = fmaf(tt, -C1, E1);             // 1 - oo
#else
                float g  = 1.0f - oo1 * fast_sigmoid(fmaf(c - mo, k1, b0_yom));
#endif
                // chain: rcp -> mul -> min -> sel -> sub -> {fma | min} -> fma
                bool  pos   = c > 0.0f;
                float ratio = u2 * __builtin_amdgcn_rcpf(c);
                float olc   = pos ? fminf(olk, ratio) : olk;
                float f     = g - olc;
                float ov    = fminf(ov1, f);
                c = fmaf(-ov, am, fmaf(f, c, u1));
            }
        };

        if (cnt == 32 && base >= tl) {
            #pragma unroll
            for (int k = 0; k < 32; ++k) step(k, false);  // branch-free fast path
        } else {
            for (int k = 0; k < cnt; ++k) step(k, true);  // boundary tiles only
        }

        if (lane < cnt) cstate[base + lane] = cs;         // coalesced store
        aC = aN; bC = bN; oC = oN;
    }
}

// ---------------------------------------------------------------------------
// Kernel C: parallel output materialization from c0[i] (written by k_scan
// into the c_n region) — re-derives gates and writes all 14*N floats.
// ---------------------------------------------------------------------------
__global__ __launch_bounds__(256) void k_out(
    const float* __restrict__ x, const float* __restrict__ ws,
    float* __restrict__ out, int n, const int* __restrict__ time_lag_p)
{
    int i = blockIdx.x * 256 + threadIdx.x;
    if (i >= n) return;
    const int tl = *time_lag_p;

    const float mo = ws[0], so = ws[1], wb1 = ws[2], wb2 = ws[3];
    const float b0_yom = ws[4], b0_ylm = ws[5], thr = ws[6];
    const float oo1 = ws[7], ol1 = ws[8], svm = ws[9], obsstd = ws[10];
    const float k1   = wb1 / so;
    const float k2   = wb2 * (1.0f / SL_C);
    const float invS = 1.0f / SCALE_MR_C;

    float c0 = out[n + i];            // raw pre-update state from k_scan
    float u2 = x[2 * i + 1];
    bool  act = (i >= tl);

    float oo    = oo1 * fast_sigmoid(fmaf(c0 - mo, k1, b0_yom));
    float ol    = ol1 * fast_sigmoid(fmaf(u2 - ML_C, k2, b0_ylm));
    bool  pos   = c0 > 0.0f;
    float ratio = u2 * __builtin_amdgcn_rcpf(c0);
    float olc   = pos ? fminf(ol, ratio) : ol;    // == ol - relu(ol - u2/c0)
    float f     = 1.0f - oo - olc;
    float d     = c0 * invS - thr;
    float sg    = (d > 0.0f) ? 1.0f : ((d < 0.0f) ? -1.0f : 0.0f);
    float ov    = fminf(svm * sg, f);             // == ov1 - relu(ov1 - f)

    float h  = act ? oo * c0 : 0.0f;
    float os = act ? obsstd : 0.0f;

    out[i]              = h;                      // h_n
    out[n + i]          = act ? c0 : 0.0f;        // c_n
    out[2 * n + i]      = act ? ol * c0 : 0.0f;   // l_n
    out[3 * n + i]      = act ? olc * c0 : 0.0f;  // lc_n
    out[4 * n + i]      = 0.0f;                   // zeros
    out[5 * n + i]      = 0.0f;                   // zeros
    out[6 * n + i]      = act ? oo : 0.0f;        // Gate_oo
    out[7 * n + i]      = act ? ol : 0.0f;        // Gate_ol
    out[8 * n + i]      = act ? olc : 0.0f;       // Gate_olc
    out[9 * n + i]      = act ? f : 0.0f;         // Gate_f
    out[10 * n + 2 * i]     = h;                  // h_nout[:,0]
    out[10 * n + 2 * i + 1] = os;                 // h_nout[:,1]
    out[12 * n + i]     = os;                     // obs_std
    out[13 * n + i]     = act ? ov : 0.0f;        // Gate_ov
}

// ---------------------------------------------------------------------------
extern "C" void kernel_launch(void* const* d_in, const int* in_sizes, int n_in,
                              void* d_out, int out_size, void* d_ws, size_t ws_size,
                              hipStream_t stream) {
    (void)n_in; (void)out_size; (void)ws_size;

    const float* x        = (const float*)d_in[0];
    const float* y        = (const float*)d_in[1];
    const float* p_mean   = (const float*)d_in[2];
    const float* p_std    = (const float*)d_in[3];
    const float* w_r_yom  = (const float*)d_in[4];
    const float* w_r_ylm  = (const float*)d_in[5];
    const float* w_r_yfm  = (const float*)d_in[6];
    const float* w_r_yvm  = (const float*)d_in[7];
    const float* b0_yom   = (const float*)d_in[8];
    const float* w_b1_yom = (const float*)d_in[9];
    const float* b0_ylm   = (const float*)d_in[10];
    const float* w_b2_ylm = (const float*)d_in[11];
    const float* b0_yrm   = (const float*)d_in[12];
    const int*   time_lag = (const int*)d_in[14];

    const int n  = in_sizes[0] / 2;   // x is (N,1,2)
    const int ny = in_sizes[1];
    float* ws  = (float*)d_ws;
    float* out = (float*)d_out;

    k_stats<<<1, 256, 0, stream>>>(y, ny, p_mean, p_std, w_r_yom, w_r_ylm,
                                   w_r_yfm, w_r_yvm, b0_yom, w_b1_yom,
                                   b0_ylm, w_b2_ylm, b0_yrm, ws);
    k_scan<<<1, 32, 0, stream>>>(x, ws, out + n, n, time_lag);

    const int blocks = (n + 255) / 256;
    k_out<<<blocks, 256, 0, stream>>>(x, ws, out, n, time_lag);
}